// DGI_24223615550076
// MI455X (gfx1250) — compile-verified
//
#include <hip/hip_runtime.h>

typedef __attribute__((ext_vector_type(2))) float v2f;
typedef __attribute__((ext_vector_type(8))) float v8f;

#define SELU_ALPHA 1.6732632423543772f
#define SELU_SCALE 1.0507009873554805f

__device__ __forceinline__ float dgi_selu(float x) {
  return x > 0.0f ? SELU_SCALE * x : SELU_SCALE * SELU_ALPHA * expm1f(x);
}

// ---------------- zero init (d_out nodes region + ws head) ----------------
__global__ void dgi_zero2(float* __restrict__ a, long long na,
                          float* __restrict__ b, long long nb) {
  long long i = (long long)blockIdx.x * blockDim.x + threadIdx.x;
  if (i < na) a[i] = 0.0f;
  if (i < nb) b[i] = 0.0f;
}

// ---------------- degree counting ----------------
__global__ void dgi_degree(const int* __restrict__ senders,
                           const int* __restrict__ receivers,
                           float* __restrict__ deg_s, float* __restrict__ deg_r, int E) {
  int e = blockIdx.x * blockDim.x + threadIdx.x;
  if (e < E) {
    atomicAdd(&deg_s[senders[e]], 1.0f);
    atomicAdd(&deg_r[receivers[e]], 1.0f);
  }
}

__global__ void dgi_invsqrt(float* __restrict__ deg_s, float* __restrict__ deg_r, int n) {
  int i = blockIdx.x * blockDim.x + threadIdx.x;
  if (i < n) {
    deg_s[i] = rsqrtf(fmaxf(deg_s[i], 1.0f));
    deg_r[i] = rsqrtf(fmaxf(deg_r[i], 1.0f));
  }
}

// ---------------- GEMM via V_WMMA_F32_16X16X4_F32 ----------------
// Each wave: 16 rows x 64 cols output strip. HS[m] = (X[m] @ W + bias) * inv_s[m]
__global__ void __launch_bounds__(256)
dgi_gemm_wmma(const float* __restrict__ X, const float* __restrict__ W,
              const float* __restrict__ bias, const float* __restrict__ inv_s,
              float* __restrict__ HS, int n) {
  const int lane = threadIdx.x & 31;
  const int wave = threadIdx.x >> 5;
  const int m0 = (blockIdx.x * 8 + wave) * 16;
  if (m0 >= n) return;                    // wave-uniform exit: EXEC all-ones for WMMA
  const int half = lane >> 4;             // 0: K even pair base 0, 1: base 2
  const int l4 = lane & 15;

  int mrow = m0 + l4;
  if (mrow >= n) mrow = n - 1;            // clamp (in-bounds garbage, store-guarded)
  const float* xrow = X + (long long)mrow * 128 + half * 2;

  v8f c0 = {}, c1 = {}, c2 = {}, c3 = {};
#pragma unroll 4
  for (int k = 0; k < 128; k += 4) {
    v2f a = *(const v2f*)(xrow + k);      // A 16x4 f32 fragment
    const float* wk = W + (k + half * 2) * 64 + l4;  // B 4x16 fragment: K in VGPR idx, +2 by half
    v2f b0 = { wk[0],  wk[64] };
    v2f b1 = { wk[16], wk[80] };
    v2f b2 = { wk[32], wk[96] };
    v2f b3 = { wk[48], wk[112] };
    c0 = __builtin_amdgcn_wmma_f32_16x16x4_f32(false, a, false, b0, (short)0, c0, false, false);
    c1 = __builtin_amdgcn_wmma_f32_16x16x4_f32(false, a, false, b1, (short)0, c1, false, false);
    c2 = __builtin_amdgcn_wmma_f32_16x16x4_f32(false, a, false, b2, (short)0, c2, false, false);
    c3 = __builtin_amdgcn_wmma_f32_16x16x4_f32(false, a, false, b3, (short)0, c3, false, false);
  }

#pragma unroll
  for (int r = 0; r < 8; ++r) {
    int m = m0 + half * 8 + r;            // C/D layout: VGPR r -> M=r (lanes 0-15), M=r+8 (16-31)
    if (m < n) {
      float is = inv_s[m];
      float* outp = HS + (long long)m * 64 + l4;
      outp[0]  = (c0[r] + bias[l4])      * is;
      outp[16] = (c1[r] + bias[16 + l4]) * is;
      outp[32] = (c2[r] + bias[32 + l4]) * is;
      outp[48] = (c3[r] + bias[48 + l4]) * is;
    }
  }
}

// ---------------- edge scatter: acc[recv] += hs[send] ----------------
__global__ void dgi_scatter(const float* __restrict__ hs, const int* __restrict__ senders,
                            const int* __restrict__ receivers, float* __restrict__ acc, int E) {
  long long idx = (long long)blockIdx.x * blockDim.x + threadIdx.x;
  int e = (int)(idx >> 6);
  int f = (int)(idx & 63);
  if (e < E) {
    atomicAdd(&acc[(long long)receivers[e] * 64 + f],
              hs[(long long)senders[e] * 64 + f]);
  }
}

// ---------------- finalize: selu(acc * inv_r) in place, optional column sum ----------------
__global__ void __launch_bounds__(256)
dgi_finalize(float* __restrict__ nodes, const float* __restrict__ inv_r,
             float* __restrict__ colsum, int n, int do_sum) {
  __shared__ float sred[256];
  const int t = threadIdx.x;
  const int f = t & 63;
  const int rq = t >> 6;                  // 0..3
  const int row0 = blockIdx.x * 64;
  float local = 0.0f;
#pragma unroll
  for (int i = 0; i < 64; i += 4) {
    int row = row0 + i + rq;
    if (row < n) {
      long long off = (long long)row * 64 + f;
      float v = dgi_selu(nodes[off] * inv_r[row]);
      nodes[off] = v;
      local += v;
    }
  }
  if (do_sum) {
    sred[t] = local;
    __syncthreads();
    if (rq == 0) {
      float s = sred[f] + sred[64 + f] + sred[128 + f] + sred[192 + f];
      atomicAdd(&colsum[f], s);
    }
  }
}

// ---------------- summary = sigmoid(colsum/N); v = W_bilinear @ summary ----------------
__global__ void dgi_summary(const float* __restrict__ colsum, const float* __restrict__ Wb,
                            float* __restrict__ summary_out, float* __restrict__ v_out, int n) {
  __shared__ float ssum[64];
  int t = threadIdx.x;                    // 64 threads
  float s = 1.0f / (1.0f + expf(-(colsum[t] / (float)n)));
  ssum[t] = s;
  summary_out[t] = s;
  __syncthreads();
  float acc = 0.0f;
#pragma unroll 8
  for (int j = 0; j < 64; ++j) acc += Wb[t * 64 + j] * ssum[j];
  v_out[t] = acc;
}

// ---------------- logits[r] = nodes[r] . v ----------------
__global__ void __launch_bounds__(256)
dgi_logits(const float* __restrict__ nodes, const float* __restrict__ v,
           float* __restrict__ logits, int total) {
  __shared__ float sm[256];
  const int t = threadIdx.x;
  const int f = t & 63;
  const int rq = t >> 6;
  const int row = blockIdx.x * 4 + rq;
  float p = 0.0f;
  if (row < total) p = nodes[(long long)row * 64 + f] * v[f];
  sm[t] = p;
  __syncthreads();
#pragma unroll
  for (int off = 32; off > 0; off >>= 1) {
    if (f < off) sm[t] += sm[t + off];
    __syncthreads();
  }
  if (f == 0 && row < total) logits[row] = sm[t];
}

extern "C" void kernel_launch(void* const* d_in, const int* in_sizes, int n_in,
                              void* d_out, int out_size, void* d_ws, size_t ws_size,
                              hipStream_t stream) {
  const float* x         = (const float*)d_in[0];
  const float* cx        = (const float*)d_in[1];
  const int*   senders   = (const int*)d_in[2];
  const int*   receivers = (const int*)d_in[3];
  const float* Wd        = (const float*)d_in[4];
  const float* bd        = (const float*)d_in[5];
  const float* Wb        = (const float*)d_in[6];

  const int N = in_sizes[0] / 128;
  const int E = in_sizes[2];

  float* out     = (float*)d_out;
  float* nodes1  = out;                              // N*64
  float* nodes2  = out + (long long)N * 64;          // N*64
  float* summary = out + (long long)2 * N * 64;      // 64
  float* logits  = summary + 64;                     // 2N

  float* ws     = (float*)d_ws;
  float* inv_s  = ws;                                // N
  float* inv_r  = ws + N;                            // N
  float* colsum = ws + 2LL * N;                      // 64
  float* vvec   = colsum + 64;                       // 64
  float* hs     = vvec + 64;                         // N*64 (reused for both passes)

  const long long nodes_elems = 2LL * N * 64;
  const long long head_elems  = 2LL * N + 128;
  dgi_zero2<<<(int)((nodes_elems + 255) / 256), 256, 0, stream>>>(out, nodes_elems, ws, head_elems);

  dgi_degree<<<(E + 255) / 256, 256, 0, stream>>>(senders, receivers, inv_s, inv_r, E);
  dgi_invsqrt<<<(N + 255) / 256, 256, 0, stream>>>(inv_s, inv_r, N);

  const int gemm_blocks = (N + 127) / 128;
  const long long sthreads = (long long)E * 64;
  const int sblocks = (int)((sthreads + 255) / 256);
  const int fblocks = (N + 63) / 64;

  // pass 1: x -> nodes1 (with column-sum for summary)
  dgi_gemm_wmma<<<gemm_blocks, 256, 0, stream>>>(x, Wd, bd, inv_s, hs, N);
  dgi_scatter<<<sblocks, 256, 0, stream>>>(hs, senders, receivers, nodes1, E);
  dgi_finalize<<<fblocks, 256, 0, stream>>>(nodes1, inv_r, colsum, N, 1);

  // pass 2: cx -> nodes2
  dgi_gemm_wmma<<<gemm_blocks, 256, 0, stream>>>(cx, Wd, bd, inv_s, hs, N);
  dgi_scatter<<<sblocks, 256, 0, stream>>>(hs, senders, receivers, nodes2, E);
  dgi_finalize<<<fblocks, 256, 0, stream>>>(nodes2, inv_r, colsum, N, 0);

  dgi_summary<<<1, 64, 0, stream>>>(colsum, Wb, summary, vvec, N);
  dgi_logits<<<(2 * N + 3) / 4, 256, 0, stream>>>(nodes1, vvec, logits, 2 * N);
}